// CvxSolver_49039936585811
// MI455X (gfx1250) — compile-verified
//
#include <hip/hip_runtime.h>

// ---------------------------------------------------------------------------
// Problem constants (fixed by the reference): B=1024, m=128, n=256, 200 iters
// ---------------------------------------------------------------------------
#define MDIM 128
#define NDIM 256
#define LDA  257          // padded LDS row stride (floats) -> conflict-free
#define ITERS 200

typedef float v2f __attribute__((ext_vector_type(2)));
typedef float v8f __attribute__((ext_vector_type(8)));

// ---------------------------------------------------------------------------
// Kernel 1: one workgroup (128 threads, 4 waves) per batch element.
// A_b (128KB) is staged into LDS once; all 200 PDHG iterations run from LDS.
// ---------------------------------------------------------------------------
__global__ __launch_bounds__(128) void pdhg_solver_kernel(
    const float* __restrict__ A,    // [B, 128, 256]
    const float* __restrict__ bvec, // [B, 128]
    const float* __restrict__ cvec, // [B, 256]
    float* __restrict__ Xout)       // [B, 256]
{
    extern __shared__ float smem[];
    float* sA    = smem;                 // [128][257]
    float* sx    = sA + MDIM * LDA;      // [256]
    float* sxbar = sx + NDIM;            // [256]
    float* sy    = sxbar + NDIM;         // [128]
    float* sb    = sy + MDIM;            // [128]
    float* sc    = sb + MDIM;            // [256]
    float* sred  = sc + NDIM;            // [128] reduction scratch

    const int tid  = threadIdx.x;
    const int bidx = blockIdx.x;
    const float* Ab = A + (size_t)bidx * (MDIM * NDIM);

    // ---- Stage A into LDS (float4 global loads), accumulate ||A||_F^2 ----
    float sumsq = 0.0f;
    const float4* A4 = (const float4*)Ab;
    for (int i4 = tid; i4 < (MDIM * NDIM) / 4; i4 += 128) {
        float4 v = A4[i4];
        int row = i4 >> 6;            // (i4*4)/256
        int col = (i4 & 63) << 2;     // (i4*4)%256
        float* dst = &sA[row * LDA + col];
        dst[0] = v.x; dst[1] = v.y; dst[2] = v.z; dst[3] = v.w;
        sumsq += v.x * v.x + v.y * v.y + v.z * v.z + v.w * v.w;
    }
    // ---- Stage vectors, init state ----
    sb[tid] = bvec[(size_t)bidx * MDIM + tid];
    sy[tid] = 0.0f;
    for (int j = tid; j < NDIM; j += 128) {
        sc[j]    = cvec[(size_t)bidx * NDIM + j];
        sx[j]    = 0.0f;
        sxbar[j] = 0.0f;
    }
    sred[tid] = sumsq;
    __syncthreads();
    for (int off = 64; off > 0; off >>= 1) {
        if (tid < off) sred[tid] += sred[tid + off];
        __syncthreads();
    }
    const float step = 0.9f / fmaxf(sqrtf(sred[0]), 1e-8f); // tau == sigma

    // ---- 200 PDHG iterations entirely out of LDS ----
    for (int it = 0; it < ITERS; ++it) {
        // y-update: thread tid owns row tid.  rowp[k] banks = (tid+k)%64.
        float acc = 0.0f;
        const float* rowp = &sA[tid * LDA];
        #pragma unroll 8
        for (int k = 0; k < NDIM; ++k)
            acc = fmaf(rowp[k], sxbar[k], acc);          // sxbar[k] broadcast
        sy[tid] = fmaxf(sy[tid] + step * (acc - sb[tid]), 0.0f); // own element
        __syncthreads();   // all sy written; all sxbar reads done

        // x-update: thread tid owns columns tid and tid+128 (share sy load)
        float acc0 = 0.0f, acc1 = 0.0f;
        #pragma unroll 8
        for (int mrow = 0; mrow < MDIM; ++mrow) {
            const float ym = sy[mrow];                   // broadcast
            const float* ap = &sA[mrow * LDA];
            acc0 = fmaf(ap[tid],       ym, acc0);        // consecutive banks
            acc1 = fmaf(ap[tid + 128], ym, acc1);
        }
        {
            const int j0 = tid, j1 = tid + 128;
            float xo0 = sx[j0], xo1 = sx[j1];
            float xn0 = fmaxf(xo0 - step * (sc[j0] + acc0), 0.0f);
            float xn1 = fmaxf(xo1 - step * (sc[j1] + acc1), 0.0f);
            sx[j0] = xn0;               sx[j1] = xn1;
            sxbar[j0] = 2.0f * xn0 - xo0;
            sxbar[j1] = 2.0f * xn1 - xo1;
        }
        __syncthreads();   // sx/sxbar ready for next iteration's reads
    }

    for (int j = tid; j < NDIM; j += 128)
        Xout[(size_t)bidx * NDIM + j] = sx[j];
}

// ---------------------------------------------------------------------------
// Kernel 2: out = X @ W^T + bias via V_WMMA_F32_16X16X4_F32.
// One wave32 per 16x16 output tile; K marched 4 at a time (64 WMMAs/tile).
// A tile: X rows (M=batch), B tile: B[k][n] = W[n][k] (W is row-major [n][k]).
// ---------------------------------------------------------------------------
__global__ __launch_bounds__(256) void epilogue_wmma_kernel(
    const float* __restrict__ X,    // [1024, 256]
    const float* __restrict__ W,    // [256, 256]
    const float* __restrict__ bias, // [256]
    float* __restrict__ out)        // [1024, 256]
{
    const int lane = threadIdx.x & 31;
    const int wave = threadIdx.x >> 5;
    const int tile = blockIdx.x * 8 + wave;   // 1024 tiles total
    const int tm = tile >> 4;                 // 64 batch tiles
    const int tn = tile & 15;                 // 16 output-dim tiles
    const int m0 = tm << 4;
    const int n0 = tn << 4;
    const int half = lane >> 4;               // K sub-pair selector
    const int l    = lane & 15;               // M (for A) / N (for B,C,D)

    // ISA layout, 32-bit A 16x4: lanes0-15 {K0,K1}, lanes16-31 {K2,K3}
    const float* xp = X + (size_t)(m0 + l) * NDIM + 2 * half;
    const float* wp = W + (size_t)(n0 + l) * NDIM + 2 * half;

    v8f acc = {};
    #pragma unroll 4
    for (int k0 = 0; k0 < NDIM; k0 += 4) {
        v2f a  = *(const v2f*)(xp + k0);   // 8B aligned global_load_b64
        v2f bm = *(const v2f*)(wp + k0);
        acc = __builtin_amdgcn_wmma_f32_16x16x4_f32(
            /*neg_a=*/false, a, /*neg_b=*/false, bm,
            /*c_mod=*/(short)0, acc, /*reuse_a=*/false, /*reuse_b=*/false);
    }

    // C/D layout: VGPR r -> M = r (lanes 0-15) or r+8 (lanes 16-31), N = l
    const float bj = bias[n0 + l];
    #pragma unroll
    for (int r = 0; r < 8; ++r) {
        const int mi = m0 + r + half * 8;
        out[(size_t)mi * NDIM + (n0 + l)] = acc[r] + bj;
    }
}

// ---------------------------------------------------------------------------
// Launch: inputs in setup_inputs() order: A, b, c, W, bias (all fp32).
// d_ws holds the intermediate solution X [1024*256 floats = 1 MB].
// ---------------------------------------------------------------------------
extern "C" void kernel_launch(void* const* d_in, const int* in_sizes, int n_in,
                              void* d_out, int out_size, void* d_ws, size_t ws_size,
                              hipStream_t stream) {
    const float* A    = (const float*)d_in[0];
    const float* b    = (const float*)d_in[1];
    const float* c    = (const float*)d_in[2];
    const float* W    = (const float*)d_in[3];
    const float* bias = (const float*)d_in[4];
    float* out  = (float*)d_out;
    float* Xbuf = (float*)d_ws;

    const size_t smem_bytes =
        (size_t)(MDIM * LDA + NDIM + NDIM + MDIM + MDIM + NDIM + MDIM) * sizeof(float);
    // ~136 KB -> 2 workgroups per 320KB WGP

    pdhg_solver_kernel<<<1024, 128, smem_bytes, stream>>>(A, b, c, Xbuf);
    epilogue_wmma_kernel<<<128, 256, 0, stream>>>(Xbuf, W, bias, out);
}